// Filter_88656714924643
// MI455X (gfx1250) — compile-verified
//
#include <hip/hip_runtime.h>

// Kalman filter, B=512 T=1024 S=64 O=32 U=16.
// Phase 1: batch-independent gain recursion (serial, early-exit at bitwise fixed point).
// Phase 2: batched state recursion as WMMA f32 16x16x4 GEMM chain.

#define KB 512
#define KT 1024
#define KS 64
#define KO 32
#define KU 16
#define TP1 1025

// LDS pitches (floats) to avoid bank conflicts
#define PA 65   // rows of width 64
#define PW 33   // rows of width 32
#define PS 33
#define PB 17   // rows of width 16

// dynamic-LDS offsets (in floats) for kf_gains
enum {
  OFF_A   = 0,
  OFF_BM  = OFF_A  + 64 * PA,          // 4160
  OFF_C   = OFF_BM + 64 * PB,          // +1088
  OFF_CA  = OFF_C  + 32 * PA,          // +2080
  OFF_CB  = OFF_CA + 32 * PA,          // +2080
  OFF_P   = OFF_CB + 32 * PB,          // +544
  OFF_PP  = OFF_P  + 64 * PA,          // +4160
  OFF_U0  = OFF_PP + 64 * PA,          // union: T1 (64*PA) | {W, S, Si}
  OFF_K   = OFF_U0 + 4224,
  OFF_FLAG= OFF_K  + 64 * PW,
  SMEM_FLOATS = OFF_FLAG + 4
};

// Per-step packed gain block: BM pairs (64x64) | BG pairs (16x64) | BK pairs (32x64)
// element (k,n) stored at (k>>1)*128 + n*2 + (k&1)  -> float2 loads give (k,k+1) at col n.
#define GAIN_STRIDE 7168   // 4096 + 1024 + 2048 floats

typedef float v2f __attribute__((ext_vector_type(2)));
typedef float v8f __attribute__((ext_vector_type(8)));

// ---------------------------------------------------------------------------
// Phase 1: gains. One workgroup, 1024 threads (32 wave32s on one WGP).
// ---------------------------------------------------------------------------
__global__ __launch_bounds__(1024, 1)
void kf_gains(const float* __restrict__ Ag, const float* __restrict__ Bg,
              const float* __restrict__ Cg, const float* __restrict__ Qg,
              const float* __restrict__ Rg,
              int* __restrict__ hdr, float* __restrict__ gains, int t_max)
{
  extern __shared__ float sm[];
  float* sA  = sm + OFF_A;
  float* sBm = sm + OFF_BM;
  float* sC  = sm + OFF_C;
  float* sCA = sm + OFF_CA;
  float* sCB = sm + OFF_CB;
  float* sP  = sm + OFF_P;
  float* sPp = sm + OFF_PP;
  float* sT1 = sm + OFF_U0;                       // live s1..s2
  float* sW  = sm + OFF_U0;                       // live s3..s7 (union w/ T1)
  float* sS  = sm + OFF_U0 + 64 * PW;
  float* sSi = sm + OFF_U0 + 64 * PW + 32 * PS;
  float* sK  = sm + OFF_K;
  int*   sFlag = (int*)(sm + OFF_FLAG);

  const int tid = threadIdx.x;

  // load constants, P = I
  for (int o = tid; o < 64 * 64; o += 1024) { int r = o >> 6, c = o & 63; sA[r * PA + c] = Ag[o]; }
  for (int o = tid; o < 64 * 16; o += 1024) { int r = o >> 4, c = o & 15; sBm[r * PB + c] = Bg[o]; }
  for (int o = tid; o < 32 * 64; o += 1024) { int r = o >> 6, c = o & 63; sC[r * PA + c] = Cg[o]; }
  for (int o = tid; o < 64 * 64; o += 1024) { int r = o >> 6, c = o & 63; sP[r * PA + c] = (r == c) ? 1.0f : 0.0f; }
  __syncthreads();

  // CA = C*A (32x64), CB = C*B (32x16) -- constant across t
  for (int o = tid; o < 32 * 64; o += 1024) {
    int i = o >> 6, j = o & 63; float acc = 0.0f;
    for (int k = 0; k < 64; ++k) acc += sC[i * PA + k] * sA[k * PA + j];
    sCA[i * PA + j] = acc;
  }
  for (int o = tid; o < 32 * 16; o += 1024) {
    int i = o >> 4, j = o & 15; float acc = 0.0f;
    for (int k = 0; k < 64; ++k) acc += sC[i * PA + k] * sBm[k * PB + j];
    sCB[i * PB + j] = acc;
  }
  __syncthreads();

  const int i0  = (tid >> 5) << 1;   // 2x2 tile row for 64x64 stages
  const int j0  = (tid & 31) << 1;
  const int i16 = tid >> 4;          // 0..63 for 64x32 stages
  const int j16 = (tid & 15) << 1;   // 0..30
  const int i32 = tid >> 5;          // 0..31 for 32x32 stage
  const int j32 = tid & 31;

  int  tconv = t_max - 1;
  bool done  = false;

  for (int t = 0; t < t_max; ++t) {
    // s1: T1 = A * P
    {
      float c00 = 0, c01 = 0, c10 = 0, c11 = 0;
      for (int k = 0; k < 64; ++k) {
        float a0 = sA[i0 * PA + k], a1 = sA[(i0 + 1) * PA + k];
        float b0 = sP[k * PA + j0], b1 = sP[k * PA + j0 + 1];
        c00 += a0 * b0; c01 += a0 * b1; c10 += a1 * b0; c11 += a1 * b1;
      }
      sT1[i0 * PA + j0] = c00;       sT1[i0 * PA + j0 + 1] = c01;
      sT1[(i0 + 1) * PA + j0] = c10; sT1[(i0 + 1) * PA + j0 + 1] = c11;
    }
    __syncthreads();
    // s2: Pp = T1 * A^T + Q
    {
      float c00 = 0, c01 = 0, c10 = 0, c11 = 0;
      for (int k = 0; k < 64; ++k) {
        float a0 = sT1[i0 * PA + k], a1 = sT1[(i0 + 1) * PA + k];
        float b0 = sA[j0 * PA + k],  b1 = sA[(j0 + 1) * PA + k];
        c00 += a0 * b0; c01 += a0 * b1; c10 += a1 * b0; c11 += a1 * b1;
      }
      sPp[i0 * PA + j0]       = c00 + Qg[i0 * 64 + j0];
      sPp[i0 * PA + j0 + 1]   = c01 + Qg[i0 * 64 + j0 + 1];
      sPp[(i0 + 1) * PA + j0]     = c10 + Qg[(i0 + 1) * 64 + j0];
      sPp[(i0 + 1) * PA + j0 + 1] = c11 + Qg[(i0 + 1) * 64 + j0 + 1];
    }
    __syncthreads();
    // s3: W = Pp * C^T  (64x32)
    {
      float w0 = 0, w1 = 0;
      for (int k = 0; k < 64; ++k) {
        float p = sPp[i16 * PA + k];
        w0 += p * sC[j16 * PA + k];
        w1 += p * sC[(j16 + 1) * PA + k];
      }
      sW[i16 * PW + j16] = w0; sW[i16 * PW + j16 + 1] = w1;
    }
    __syncthreads();
    // s4: S = C * W + R  (32x32)
    {
      float acc = 0.0f;
      for (int k = 0; k < 64; ++k) acc += sC[i32 * PA + k] * sW[k * PW + j32];
      sS[i32 * PS + j32] = acc + Rg[i32 * 32 + j32];
    }
    __syncthreads();
    // s5: Si = inv(S) -- single wave32, column-per-lane Gauss-Jordan, all in VGPRs
    if (tid < 32) {
      const int j = tid;
      float m[32], vi[32];
      #pragma unroll
      for (int i = 0; i < 32; ++i) { m[i] = sS[i * PS + j]; vi[i] = (i == j) ? 1.0f : 0.0f; }
      #pragma unroll
      for (int kk = 0; kk < 32; ++kk) {
        float c[32];
        #pragma unroll
        for (int i = 0; i < 32; ++i) c[i] = __shfl(m[i], kk, 32);  // column kk broadcast
        const float ip = 1.0f / c[kk];
        m[kk] *= ip; vi[kk] *= ip;
        #pragma unroll
        for (int i = 0; i < 32; ++i) {
          if (i == kk) continue;
          m[i]  -= c[i] * m[kk];
          vi[i] -= c[i] * vi[kk];
        }
      }
      #pragma unroll
      for (int i = 0; i < 32; ++i) sSi[i * PS + j] = vi[i];
    }
    __syncthreads();
    // s6: K = W * Si  (64x32); reset compare flag
    {
      float k0 = 0, k1 = 0;
      for (int k = 0; k < 32; ++k) {
        float w = sW[i16 * PW + k];
        k0 += w * sSi[k * PS + j16];
        k1 += w * sSi[k * PS + j16 + 1];
      }
      sK[i16 * PW + j16] = k0; sK[i16 * PW + j16 + 1] = k1;
      if (tid == 0) *sFlag = 0;
    }
    __syncthreads();
    // s7: Pnew = Pp - K * W^T ; bitwise compare with previous P
    {
      float c00 = sPp[i0 * PA + j0],       c01 = sPp[i0 * PA + j0 + 1];
      float c10 = sPp[(i0 + 1) * PA + j0], c11 = sPp[(i0 + 1) * PA + j0 + 1];
      for (int k = 0; k < 32; ++k) {
        float a0 = sK[i0 * PW + k], a1 = sK[(i0 + 1) * PW + k];
        float b0 = sW[j0 * PW + k], b1 = sW[(j0 + 1) * PW + k];
        c00 -= a0 * b0; c01 -= a0 * b1; c10 -= a1 * b0; c11 -= a1 * b1;
      }
      float o00 = sP[i0 * PA + j0],       o01 = sP[i0 * PA + j0 + 1];
      float o10 = sP[(i0 + 1) * PA + j0], o11 = sP[(i0 + 1) * PA + j0 + 1];
      if (__float_as_uint(c00) != __float_as_uint(o00) ||
          __float_as_uint(c01) != __float_as_uint(o01) ||
          __float_as_uint(c10) != __float_as_uint(o10) ||
          __float_as_uint(c11) != __float_as_uint(o11)) *sFlag = 1;
      sP[i0 * PA + j0] = c00;       sP[i0 * PA + j0 + 1] = c01;
      sP[(i0 + 1) * PA + j0] = c10; sP[(i0 + 1) * PA + j0 + 1] = c11;
    }
    __syncthreads();
    // s8: emit packed gains for step t
    {
      float* gm = gains + (size_t)t * GAIN_STRIDE;
      float* gg = gm + 4096;
      float* gk = gm + 5120;
      const int n   = (tid & 127) >> 1;       // output column 0..63
      const int bit = tid & 1;
      const int kb  = ((tid >> 7) << 1) + bit; // base k 0..15
      float kn[32];
      #pragma unroll
      for (int m2 = 0; m2 < 32; ++m2) kn[m2] = sK[n * PW + m2];
      // BM[k][n] = A[n][k] - sum_m K[n][m]*CA[m][k]
      #pragma unroll
      for (int p = 0; p < 4; ++p) {
        int k = kb + 16 * p;
        float acc = sA[n * PA + k];
        for (int m2 = 0; m2 < 32; ++m2) acc -= kn[m2] * sCA[m2 * PA + k];
        gm[tid + (p << 10)] = acc;
      }
      // BG[k][n] = Bm[n][k] - sum_m K[n][m]*CB[m][k]   (k = kb, 0..15)
      {
        float acc = sBm[n * PB + kb];
        for (int m2 = 0; m2 < 32; ++m2) acc -= kn[m2] * sCB[m2 * PB + kb];
        gg[tid] = acc;
      }
      // BK[k][n] = K[n][k]
      #pragma unroll
      for (int p = 0; p < 2; ++p) {
        int k = kb + 16 * p;
        gk[tid + (p << 10)] = sK[n * PW + k];
      }
    }
    // bitwise fixed point -> all later gains identical; stop (exact)
    if (*sFlag == 0) { tconv = t; done = true; break; }
  }
  if (tid == 0) hdr[0] = done ? tconv : (t_max - 1);
}

// ---------------------------------------------------------------------------
// Phase 2: batched state recursion with V_WMMA_F32_16X16X4_F32.
// 32 blocks x 128 threads; block = 16 batch rows; wave w = 16-col tile.
// ---------------------------------------------------------------------------
__global__ __launch_bounds__(128, 1)
void kf_states(const float* __restrict__ obs, const float* __restrict__ inp,
               const float* __restrict__ x0, const int* __restrict__ hdr,
               const float* __restrict__ gains, float* __restrict__ out)
{
  __shared__ float xs[2][16 * 68];   // double-buffered X tile, pitch 68
  const int tid  = threadIdx.x;
  const int lane = tid & 31;
  const int wave = tid >> 5;
  const int n0   = wave << 4;
  const int b0   = blockIdx.x << 4;
  const int rm   = lane & 15;        // tile row / B-col within 16
  const int hi   = lane >> 4;        // lane half selects K pair

  // X_0 = x0 broadcast; out[:,0,:] = x0
  for (int o = tid; o < 16 * 64; o += 128) {
    int r = o >> 6, c = o & 63;
    float v = x0[c];
    xs[0][r * 68 + c] = v;
    out[(size_t)(b0 + r) * (TP1 * 64) + c] = v;
  }
  __syncthreads();
  const int tconv = hdr[0];

  const float* zrow = obs + (size_t)(b0 + rm) * (KT * KO);
  const float* urow = inp + (size_t)(b0 + rm) * (KT * KU);
  const int bcol = (n0 + rm) << 1;
  int cur = 0;

  for (int t = 0; t < KT; ++t) {
    const int idx = (t < tconv) ? t : tconv;
    const float* g  = gains + (size_t)idx * GAIN_STRIDE;
    const float* gu = g + 4096;
    const float* gk = g + 5120;

    v8f acc0 = {0.f, 0.f, 0.f, 0.f, 0.f, 0.f, 0.f, 0.f};
    v8f acc1 = {0.f, 0.f, 0.f, 0.f, 0.f, 0.f, 0.f, 0.f};

    // X * M   (K = 64 -> 16 wmma, split over 2 accumulator chains)
    const float* xr = &xs[cur][rm * 68];
    #pragma unroll
    for (int q = 0; q < 16; ++q) {
      v2f a = *(const v2f*)(xr + (q << 2) + (hi << 1));
      v2f b = *(const v2f*)(g + ((q << 1) + hi) * 128 + bcol);
      if (q & 1)
        acc1 = __builtin_amdgcn_wmma_f32_16x16x4_f32(false, a, false, b, (short)0, acc1, false, false);
      else
        acc0 = __builtin_amdgcn_wmma_f32_16x16x4_f32(false, a, false, b, (short)0, acc0, false, false);
    }
    // U_t * G   (K = 16 -> 4 wmma)
    const float* up = urow + (t << 4);
    #pragma unroll
    for (int q = 0; q < 4; ++q) {
      v2f a = *(const v2f*)(up + (q << 2) + (hi << 1));
      v2f b = *(const v2f*)(gu + ((q << 1) + hi) * 128 + bcol);
      if (q & 1)
        acc1 = __builtin_amdgcn_wmma_f32_16x16x4_f32(false, a, false, b, (short)0, acc1, false, false);
      else
        acc0 = __builtin_amdgcn_wmma_f32_16x16x4_f32(false, a, false, b, (short)0, acc0, false, false);
    }
    // Z_t * K^T (K = 32 -> 8 wmma)
    const float* zp = zrow + (t << 5);
    #pragma unroll
    for (int q = 0; q < 8; ++q) {
      v2f a = *(const v2f*)(zp + (q << 2) + (hi << 1));
      v2f b = *(const v2f*)(gk + ((q << 1) + hi) * 128 + bcol);
      if (q & 1)
        acc1 = __builtin_amdgcn_wmma_f32_16x16x4_f32(false, a, false, b, (short)0, acc1, false, false);
      else
        acc0 = __builtin_amdgcn_wmma_f32_16x16x4_f32(false, a, false, b, (short)0, acc0, false, false);
    }
    v8f acc = acc0 + acc1;

    // hint-prefetch next step's gain block (global_prefetch_b8)
    const int idn = ((t + 1) < tconv) ? (t + 1) : tconv;
    __builtin_prefetch(gains + (size_t)idn * GAIN_STRIDE + tid * 56, 0, 1);

    // write out[:, t+1, :] and next X tile (D layout: reg v -> row v + 8*hi, col n0+rm)
    const int nxt = cur ^ 1;
    const size_t tb = (size_t)(t + 1) * 64 + (n0 + rm);
    #pragma unroll
    for (int v = 0; v < 8; ++v) {
      int r = v + (hi << 3);
      out[(size_t)(b0 + r) * (TP1 * 64) + tb] = acc[v];
      xs[nxt][r * 68 + n0 + rm] = acc[v];
    }
    __syncthreads();
    cur = nxt;
  }
}

// ---------------------------------------------------------------------------
extern "C" void kernel_launch(void* const* d_in, const int* in_sizes, int n_in,
                              void* d_out, int out_size, void* d_ws, size_t ws_size,
                              hipStream_t stream) {
  (void)in_sizes; (void)n_in; (void)out_size;
  const float* obs = (const float*)d_in[0];
  const float* inp = (const float*)d_in[1];
  const float* A   = (const float*)d_in[2];
  const float* Bm  = (const float*)d_in[3];
  const float* C   = (const float*)d_in[4];
  const float* Q   = (const float*)d_in[5];
  const float* R   = (const float*)d_in[6];
  const float* x0  = (const float*)d_in[7];
  float* out   = (float*)d_out;
  int*   hdr   = (int*)d_ws;
  float* gains = (float*)d_ws + 64;

  long cap = (long)(ws_size / 4) - 64;
  int t_max = (int)(cap / GAIN_STRIDE);
  if (t_max > KT) t_max = KT;
  if (t_max < 1)  t_max = 1;

  size_t smem = (size_t)SMEM_FLOATS * sizeof(float);
  kf_gains<<<1, 1024, smem, stream>>>(A, Bm, C, Q, R, hdr, gains, t_max);
  kf_states<<<dim3(KB / 16), 128, 0, stream>>>(obs, inp, x0, hdr, gains, out);
}